// DKD_22651657519671
// MI455X (gfx1250) — compile-verified
//
#include <hip/hip_runtime.h>
#include <stdint.h>

// ---- problem constants (match reference) ----
#define H      2048
#define W      2048
#define NT     262144      // (2048/4)^2 tiles
#define TOPK   1000
#define NBINS  2048
#define CAP    4096        // candidate buffer / bitonic sort size
#define SORTN  4096
#define CH     64          // descriptor channels
#define PLANE  ((size_t)H * (size_t)W)

typedef float v2f __attribute__((ext_vector_type(2)));
typedef float v8f __attribute__((ext_vector_type(8)));

// ---------------- pass 1: per-4x4-tile max (with border zeroing) ----------------
__global__ void k_tilemax(const float* __restrict__ s,
                          float* __restrict__ tv, uint32_t* __restrict__ tp) {
    int t = blockIdx.x * blockDim.x + threadIdx.x;
    if (t >= NT) return;
    int ty = t >> 9, tx = t & 511;           // 512 tiles per row
    int r0 = ty << 2, c0 = tx << 2;
    float best = -1.0f; int bk = 0;
#pragma unroll
    for (int kr = 0; kr < 4; ++kr) {
        int r = r0 + kr;
        float4 v = *reinterpret_cast<const float4*>(s + (size_t)r * W + c0);
        float e[4] = {v.x, v.y, v.z, v.w};
        bool rbad = (r < 3) || (r >= H - 2);
#pragma unroll
        for (int kc = 0; kc < 4; ++kc) {
            int c = c0 + kc;
            float x = (rbad || (c < 3) || (c >= W - 2)) ? 0.0f : e[kc];
            if (x > best) { best = x; bk = kr * 4 + kc; }   // first-max, like jnp.argmax
        }
    }
    tv[t] = best;
    tp[t] = (uint32_t)((r0 + (bk >> 2)) * W + (c0 + (bk & 3)));  // fits in 22 bits
}

// ---------------- pass 2: histogram-based threshold for top-K ----------------
__device__ __forceinline__ int binOf(float v) {
    int b = (int)(v * (float)NBINS);
    return b < 0 ? 0 : (b > NBINS - 1 ? NBINS - 1 : b);
}

__global__ void k_init(uint32_t* hist, uint32_t* cnts) {
    int i = blockIdx.x * blockDim.x + threadIdx.x;
    if (i < NBINS) hist[i] = 0u;
    if (i < 2)     cnts[i] = 0u;     // cnts[0]=nCand, cnts[1]=threshold bin
}

__global__ void k_hist(const float* __restrict__ tv, uint32_t* __restrict__ hist) {
    int i = blockIdx.x * blockDim.x + threadIdx.x;
    if (i >= NT) return;
    atomicAdd(&hist[binOf(tv[i])], 1u);
}

__global__ void k_thresh(const uint32_t* __restrict__ hist, uint32_t* cnts) {
    uint32_t acc = 0; int T = 0;
    for (int b = NBINS - 1; b >= 0; --b) {
        acc += hist[b];
        if (acc >= TOPK) { T = b; break; }
    }
    cnts[1] = (uint32_t)T;
}

__global__ void k_compact(const float* __restrict__ tv, const uint32_t* __restrict__ tp,
                          uint32_t* cnts, uint64_t* __restrict__ cand) {
    int i = blockIdx.x * blockDim.x + threadIdx.x;
    if (i >= NT) return;
    float v = tv[i];
    if (binOf(v) >= (int)cnts[1]) {
        uint32_t idx = atomicAdd(&cnts[0], 1u);
        if (idx < CAP)
            cand[idx] = ((uint64_t)__float_as_uint(v) << 32) | (uint64_t)tp[i];
    }
}

// ---------------- pass 3: one-workgroup bitonic sort + emit kpts/scores ----------------
__global__ void __launch_bounds__(1024)
k_sort_out(const uint64_t* __restrict__ cand, const uint32_t* __restrict__ cnts,
           float* __restrict__ out, int* __restrict__ selr, int* __restrict__ selc) {
    __shared__ uint64_t key[SORTN];
    int n = (int)cnts[0]; if (n > CAP) n = CAP;
    for (int i = threadIdx.x; i < SORTN; i += blockDim.x)
        key[i] = (i < n) ? cand[i] : 0ull;           // pad with smallest key
    __syncthreads();
    for (int k = 2; k <= SORTN; k <<= 1) {
        for (int j = k >> 1; j > 0; j >>= 1) {
            for (int i = threadIdx.x; i < SORTN; i += blockDim.x) {
                int ixj = i ^ j;
                if (ixj > i) {
                    uint64_t a = key[i], b = key[ixj];
                    bool up = ((i & k) == 0);
                    if ((a > b) == up) { key[i] = b; key[ixj] = a; }
                }
            }
            __syncthreads();
        }
    }
    // ascending sort: last TOPK entries, in ascending order == reference ordering
    if (threadIdx.x < TOPK) {
        int j = threadIdx.x;
        uint64_t kk = key[SORTN - TOPK + j];
        uint32_t pos = (uint32_t)(kk & 0x3FFFFFull);
        float    val = __uint_as_float((uint32_t)(kk >> 32));
        int row = (int)(pos >> 11), col = (int)(pos & (W - 1));
        out[2 * j + 0]        = (float)col;          // kpts[:,0] = col
        out[2 * j + 1]        = (float)row;          // kpts[:,1] = row
        out[2000 + 64000 + j] = val;                 // kptscores
        selc[j] = col; selr[j] = row;
    }
}

// ---------------- pass 4: descriptor gather + WMMA-based L2 normalize ----------------
// One wave32 per 16 keypoints. Gathered values are held in the exact WMMA f32
// 16x4 A-matrix layout; D = (A .* A) x Ones accumulates per-row sum-of-squares
// into the f32 16x16 accumulator via v_wmma_f32_16x16x4_f32 (16 chained WMMAs).
__global__ void __launch_bounds__(32)
k_desc(const float* __restrict__ dmap, const int* __restrict__ selr,
       const int* __restrict__ selc, float* __restrict__ out) {
    const int lane = threadIdx.x;            // 0..31
    const int i    = lane & 15;              // keypoint slot within the 16-batch
    int m  = blockIdx.x * 16 + i;
    int mc = m > (TOPK - 1) ? (TOPK - 1) : m;    // branchless clamp: EXEC stays all-1s
    const int r = selr[mc], c = selc[mc];
    const size_t base = (size_t)r * W + c;
    const int koff = (lane >> 4) << 1;       // lanes 0-15: K=0,1 ; lanes 16-31: K=2,3

    v8f acc = {};
    const v2f ones = {1.0f, 1.0f};
    float raw[32];
#pragma unroll
    for (int kk = 0; kk < 16; ++kk) {
        int ch = kk * 4 + koff;
        float a0 = dmap[(size_t)ch       * PLANE + base];
        float a1 = dmap[(size_t)(ch + 1) * PLANE + base];
        raw[2 * kk]     = a0;
        raw[2 * kk + 1] = a1;
        v2f asq = {a0 * a0, a1 * a1};
        // D[m][n] += sum_k asq[m][k] * 1  -> squared L2 norm per keypoint row
        acc = __builtin_amdgcn_wmma_f32_16x16x4_f32(
            false, asq, false, ones, (short)0, acc, false, false);
    }

    // C/D layout: VGPR q on lanes 0-15 holds M=q; on lanes 16-31 holds M=8+q.
    __shared__ float nrm[16];
    if (lane == 0) {
#pragma unroll
        for (int q = 0; q < 8; ++q) nrm[q] = acc[q];
    }
    if (lane == 16) {
#pragma unroll
        for (int q = 0; q < 8; ++q) nrm[8 + q] = acc[q];
    }
    __syncthreads();

    float inv = 1.0f / sqrtf(nrm[i]);
    if (m < TOPK) {
        float* o = out + 2000 + (size_t)m * CH;   // descriptors[m, ch, 0]
#pragma unroll
        for (int kk = 0; kk < 16; ++kk) {
            int ch = kk * 4 + koff;
            o[ch]     = raw[2 * kk]     * inv;
            o[ch + 1] = raw[2 * kk + 1] * inv;
        }
    }
}

// ---------------- host launcher ----------------
extern "C" void kernel_launch(void* const* d_in, const int* in_sizes, int n_in,
                              void* d_out, int out_size, void* d_ws, size_t ws_size,
                              hipStream_t stream) {
    (void)in_sizes; (void)n_in; (void)out_size; (void)ws_size;
    const float* scores = (const float*)d_in[0];   // (1,1,2048,2048) f32
    const float* dmap   = (const float*)d_in[1];   // (1,64,2048,2048) f32
    float* out = (float*)d_out;                    // 2000 + 64000 + 1000 floats
    char* ws = (char*)d_ws;

    float*    tv   = (float*)   (ws);                        // 1 MB
    uint32_t* tp   = (uint32_t*)(ws + 1048576);              // 1 MB
    uint32_t* hist = (uint32_t*)(ws + 2097152);              // 8 KB
    uint32_t* cnts = (uint32_t*)(ws + 2097152 + 8192);       // 8 B
    uint64_t* cand = (uint64_t*)(ws + 2097152 + 8192 + 8);   // 32 KB (8-aligned)
    int*      selc = (int*)     (ws + 2097152 + 8192 + 8 + 32768);
    int*      selr = selc + TOPK;

    k_init   <<<(NBINS + 255) / 256, 256, 0, stream>>>(hist, cnts);
    k_tilemax<<<NT / 256,            256, 0, stream>>>(scores, tv, tp);
    k_hist   <<<NT / 256,            256, 0, stream>>>(tv, hist);
    k_thresh <<<1,                     1, 0, stream>>>(hist, cnts);
    k_compact<<<NT / 256,            256, 0, stream>>>(tv, tp, cnts, cand);
    k_sort_out<<<1,                 1024, 0, stream>>>(cand, cnts, out, selr, selc);
    k_desc   <<<(TOPK + 15) / 16,     32, 0, stream>>>(dmap, selr, selc, out);
}